// GCN_60679297958521
// MI455X (gfx1250) — compile-verified
//
#include <hip/hip_runtime.h>

typedef __attribute__((ext_vector_type(2))) float v2f;
typedef __attribute__((ext_vector_type(8))) float v8f;

// ---------------------------------------------------------------------------
// C[M,N] = A[M,K] * B[K,N], full fp32 via V_WMMA_F32_16X16X4_F32.
// One wave computes a full 16-row x N-col strip (NT = N/16 accumulator tiles,
// 8*NT VGPRs). Per k-step: 1 b64 A-load feeds NT WMMAs -> A (the 102MB node
// matrix) is read exactly once from memory; B (<=128KB) stays L1/L2 resident.
// Layouts (CDNA5 ISA 7.12.2, wave32):
//   A 16x4 : lane L -> M = L&15, VGPR v -> K = 2*(L>>4)+v   (contiguous b64)
//   B 4x16 : lane L -> N = L&15, VGPR v -> K = 2*(L>>4)+v
//   C 16x16: lane L -> N = L&15, VGPR v -> M = v + 8*(L>>4)
// Requires M % 16 == 0, K % 4 == 0, N = NT*16 (true here).
// ---------------------------------------------------------------------------
template <int NT>
__global__ __launch_bounds__(64)
void gemm_wmma_f32(const float* __restrict__ A, const float* __restrict__ B,
                   float* __restrict__ C, int M, int K) {
  constexpr int N = NT * 16;
  const int wave = threadIdx.x >> 5;
  const int lane = threadIdx.x & 31;
  const int row  = lane & 15;
  const int half = lane >> 4;
  const int mbase = (blockIdx.x * 2 + wave) * 16;
  if (mbase >= M) return;

  const float* aptr  = A + (size_t)(mbase + row) * K + half * 2;
  const float* bbase = B + (size_t)(half * 2) * N + row;

  v8f acc[NT] = {};
  for (int k = 0; k < K; k += 4) {
    const v2f a = *(const v2f*)(aptr + k);       // A[row][k+2*half .. +1]
    const float* bk = bbase + (size_t)k * N;
#pragma unroll
    for (int t = 0; t < NT; ++t) {
      v2f b;
      b.x = bk[t * 16];                          // B[k+2*half  ][16t+row]
      b.y = bk[t * 16 + N];                      // B[k+2*half+1][16t+row]
      acc[t] = __builtin_amdgcn_wmma_f32_16x16x4_f32(
          /*neg_a=*/false, a, /*neg_b=*/false, b,
          /*c_mod=*/(short)0, acc[t], /*reuse_a=*/false, /*reuse_b=*/false);
    }
  }

  float* cbase = C + (size_t)(mbase + half * 8) * N + row;
#pragma unroll
  for (int t = 0; t < NT; ++t) {
#pragma unroll
    for (int v = 0; v < 8; ++v)
      cbase[(size_t)v * N + t * 16] = acc[t][v];
  }
}

// ---------------------------------------------------------------------------
// Edge scatter: agg[dst[e], :] += sup[src[e], :] * w[e]
// D/4 threads cooperate on one edge; float4 gather (full 512B/256B row per
// wave, perfectly coalesced) + 4x global_atomic_add_f32. The 51MB/25MB agg
// buffers are L2-resident on MI455X (192MB L2), so atomics stay on-chip.
// ---------------------------------------------------------------------------
template <int D>
__global__ __launch_bounds__(256)
void scatter_add(const float* __restrict__ sup, const int* __restrict__ src,
                 const int* __restrict__ dst, const float* __restrict__ w,
                 float* __restrict__ agg, int E) {
  constexpr int TPE = D / 4;
  size_t gid = (size_t)blockIdx.x * blockDim.x + threadIdx.x;
  int e = (int)(gid / TPE);
  int f = (int)(gid % TPE) * 4;
  if (e >= E) return;
  const int s = src[e];
  const int d = dst[e];
  const float ww = w[e];
  const float4 v = *(const float4*)(sup + (size_t)s * D + f);
  float* p = agg + (size_t)d * D + f;
  atomicAdd(p + 0, v.x * ww);
  atomicAdd(p + 1, v.y * ww);
  atomicAdd(p + 2, v.z * ww);
  atomicAdd(p + 3, v.w * ww);
}

// h = relu(agg + b1) * ((drop_u > 0.5) ? 2 : 0), in place, D = 128.
__global__ __launch_bounds__(256)
void bias_relu_drop128(float* __restrict__ h, const float* __restrict__ b,
                       const float* __restrict__ du, int n_nodes) {
  size_t gid = (size_t)blockIdx.x * blockDim.x + threadIdx.x;
  size_t total = (size_t)n_nodes * 32;  // groups of 4 floats
  if (gid >= total) return;
  size_t off = gid * 4;
  int col = (int)(off & 127);
  float4 v  = *(const float4*)(h + off);
  float4 u  = *(const float4*)(du + off);
  float4 bb = *(const float4*)(b + col);
  v.x = fmaxf(v.x + bb.x, 0.0f) * (u.x > 0.5f ? 2.0f : 0.0f);
  v.y = fmaxf(v.y + bb.y, 0.0f) * (u.y > 0.5f ? 2.0f : 0.0f);
  v.z = fmaxf(v.z + bb.z, 0.0f) * (u.z > 0.5f ? 2.0f : 0.0f);
  v.w = fmaxf(v.w + bb.w, 0.0f) * (u.w > 0.5f ? 2.0f : 0.0f);
  *(float4*)(h + off) = v;
}

// out[i, :] = b2[:]  (bias pre-seeded so scatter2 can atomically accumulate)
__global__ __launch_bounds__(256)
void init_out64(float* __restrict__ out, const float* __restrict__ b, int n_nodes) {
  size_t gid = (size_t)blockIdx.x * blockDim.x + threadIdx.x;
  size_t total = (size_t)n_nodes * 16;  // groups of 4 floats
  if (gid >= total) return;
  size_t off = gid * 4;
  int col = (int)(off & 63);
  *(float4*)(out + off) = *(const float4*)(b + col);
}

extern "C" void kernel_launch(void* const* d_in, const int* in_sizes, int n_in,
                              void* d_out, int out_size, void* d_ws, size_t ws_size,
                              hipStream_t stream) {
  const float* x    = (const float*)d_in[0];  // [N, 256]
  const int*   esrc = (const int*)  d_in[1];  // [E]
  const int*   edst = (const int*)  d_in[2];  // [E]
  const float* ew   = (const float*)d_in[3];  // [E]
  const float* W1   = (const float*)d_in[4];  // [256, 128]
  const float* b1   = (const float*)d_in[5];  // [128]
  const float* W2   = (const float*)d_in[6];  // [128, 64]
  const float* b2   = (const float*)d_in[7];  // [64]
  const float* du   = (const float*)d_in[8];  // [N, 128]
  float* out = (float*)d_out;                 // [N, 64]

  const int Nn = in_sizes[0] / 256;  // 100000 (multiple of 16)
  const int E  = in_sizes[1];        // 1600000

  // Workspace: sup1 [N,128] | agg1/h [N,128] | sup2 [N,64]  (~128 MB)
  float* sup1 = (float*)d_ws;
  float* agg1 = sup1 + (size_t)Nn * 128;
  float* sup2 = agg1 + (size_t)Nn * 128;

  const int strips = Nn / 16;                 // 6250 16-row strips
  const unsigned gemm_blocks = (unsigned)((strips + 1) / 2);  // 2 waves/block

  // Layer 1: sup1 = x @ W1  (NT = 128/16 = 8 tiles per wave)
  gemm_wmma_f32<8><<<gemm_blocks, 64, 0, stream>>>(x, W1, sup1, Nn, 256);

  // agg1 = 0 (memset node is graph-capture legal)
  hipMemsetAsync(agg1, 0, (size_t)Nn * 128 * sizeof(float), stream);

  // agg1[dst] += sup1[src] * w
  {
    size_t threads = (size_t)E * 32;  // 128/4 lanes per edge
    scatter_add<128><<<(unsigned)((threads + 255) / 256), 256, 0, stream>>>(
        sup1, esrc, edst, ew, agg1, E);
  }

  // h = relu(agg1 + b1) * dropout  (in place)
  {
    size_t threads = (size_t)Nn * 32;
    bias_relu_drop128<<<(unsigned)((threads + 255) / 256), 256, 0, stream>>>(
        agg1, b1, du, Nn);
  }

  // Layer 2: sup2 = h @ W2  (NT = 64/16 = 4 tiles per wave)
  gemm_wmma_f32<4><<<gemm_blocks, 64, 0, stream>>>(agg1, W2, sup2, Nn, 128);

  // out = b2 (broadcast), then out[dst] += sup2[src] * w
  {
    size_t threads = (size_t)Nn * 16;
    init_out64<<<(unsigned)((threads + 255) / 256), 256, 0, stream>>>(out, b2, Nn);
  }
  {
    size_t threads = (size_t)E * 16;  // 64/4 lanes per edge
    scatter_add<64><<<(unsigned)((threads + 255) / 256), 256, 0, stream>>>(
        sup2, esrc, edst, ew, out, E);
  }
}